// DualInhibitionMemoryLayer_13134009991921
// MI455X (gfx1250) — compile-verified
//
#include <hip/hip_runtime.h>

// =====================================================================
// DualInhibitionMemoryLayer on gfx1250 (MI455X)
//
// Pipeline:
//   K1 proj_gemm : i = x @ W^T           (8192x512x128 f32 GEMM, WMMA f32 16x16x4,
//                                         A-tile staged via TDM tensor_load_to_lds)
//   K2 key_scan  : keys/key_traces for all t (pure EMA scan, no recurrence on mem)
//   K3 scan      : chunked recurrence, mem resident in LDS (266KB padded),
//                  per 16-step chunk:
//                    key/kt chunks staged via TDM (pad feature -> stride 260)
//                    M    = mem @ Keys_chunk      (WMMA, 256x16x256)
//                    G    = KT_chunk^T @ Keys     (WMMA, 16x16x256 Gram)
//                    16 sequential VALU steps using M, G (corrections)
//                    mem += LR * VT @ KT^T        (WMMA rank-16 update, C-accum)
//
// Workspace: iProj 16MB + ktBuf 8MB = 24MB in d_ws.
// Output: [mem (B*H*H) | keys (B*T*H) | vals (B*T*H)] f32.
// =====================================================================

typedef float v2f __attribute__((ext_vector_type(2)));
typedef float v8f __attribute__((ext_vector_type(8)));
typedef unsigned int u32x4 __attribute__((ext_vector_type(4)));
typedef int i32x4 __attribute__((ext_vector_type(4)));
typedef int i32x8 __attribute__((ext_vector_type(8)));

#define ALPHA_C   0.95122942450071403f   // exp(-1/20)
#define OMA_C     0.04877057549928597f   // 1 - exp(-1/20)
#define DT_C      0.95122942450071403f   // trace decay (same tau)
#define OMD_C     0.04877057549928597f
#define LR_C      0.001f
#define RECALL_C  0.2f

#define Bsz  64
#define Tsz  128
#define Hsz  256
#define INsz 128

#if defined(__HIP_DEVICE_COMPILE__) && __has_builtin(__builtin_amdgcn_tensor_load_to_lds)
#define HAVE_TDM 1
#else
#define HAVE_TDM 0
#endif

// D = A(16x4) * B(4x16) + C  in fp32 (exact-precision WMMA path on CDNA5)
static __device__ __forceinline__ v8f wmma_k4(v2f a, v2f b, v8f c) {
  return __builtin_amdgcn_wmma_f32_16x16x4_f32(false, a, false, b, (short)0, c,
                                               false, false);
}

#if HAVE_TDM
// TDM 2-D tile load: rows x rowElems f32 tile from a row-major tensor
// (row stride = strideElems) into LDS at lds_byte_off, inserting
// (padAmtCode+1) DWORDs of LDS padding every 2^(padIntCode+1) DWORDs.
// D# built per CDNA5 ISA (groups 2/3 = 0, 2-D).
// This toolchain exposes the 6-arg builtin:
//   (uint32x4 g0, int32x8 g1, int32x4 g2, int32x4 g3, int32x8 extra, i32 cpol)
static __device__ __forceinline__ void tdm_load_2d(const void* gsrc,
                                                   unsigned lds_byte_off,
                                                   int rowElems, int rows,
                                                   int strideElems,
                                                   int padIntCode,
                                                   int padAmtCode) {
  const unsigned long long ga = (unsigned long long)(__SIZE_TYPE__)gsrc;
  u32x4 g0;
  g0[0] = 1u;                                   // count=1 (valid user D#)
  g0[1] = lds_byte_off;                         // lds_addr (bytes)
  g0[2] = (unsigned)(ga & 0xffffffffu);         // global_addr[31:0]
  g0[3] = (unsigned)((ga >> 32) & 0x01ffffffu)  // global_addr[56:32]
          | (2u << 30);                         // type=2 ("image")
  i32x8 g1;
  g1[0] = (2 << 16)                     // data_size = 4 bytes
          | (1 << 20)                   // pad_enable
          | (padIntCode << 22)          // pad_interval
          | (padAmtCode << 25);         // pad_amount
  g1[1] = (rowElems & 0xffff) << 16;    // [47:32]=abar=0, [63:48]=tensor_dim0 lo
  g1[2] = ((unsigned)rowElems >> 16) |  // tensor_dim0 hi16
          ((rows & 0xffff) << 16);      // tensor_dim1 lo16 (= tile rows)
  g1[3] = (rowElems & 0xffff) << 16;    // tensor_dim1 hi=0, tile_dim0
  g1[4] = rows & 0xffff;                // tile_dim1, tile_dim2=0
  g1[5] = strideElems;                  // tensor_dim0_stride lo32
  g1[6] = 0;                            // stride hi16, tensor_dim1_stride lo16
  g1[7] = 0;                            // tensor_dim1_stride hi
  const i32x4 z4 = {0, 0, 0, 0};
  const i32x8 z8 = {0, 0, 0, 0, 0, 0, 0, 0};
  __builtin_amdgcn_tensor_load_to_lds(g0, g1, z4, z4, z8, 0);
}
#endif

// ---------------------------------------------------------------------
// K1: iProj[(b*T+t)][hh] = sum_k x[b,t,k] * W[hh,k]   (M=8192,N=512,K=128)
// grid = (512 M-tiles, 4 N-groups), block = 256 (8 waves, 1 N-tile each)
// ---------------------------------------------------------------------
__global__ __launch_bounds__(256) void proj_gemm(const float* __restrict__ x,
                                                 const float* __restrict__ W,
                                                 float* __restrict__ iProj) {
  __shared__ float xT[16 * 132];  // 16 rows x 128 K, pad 132 (bank-friendly)
  const int tid = threadIdx.x, lane = tid & 31, w = tid >> 5;
  const int lm = lane & 15, hl = lane >> 4, cs = hl * 2;
  const int m0 = blockIdx.x * 16;
  const int n0 = (blockIdx.y * 8 + w) * 16;

#if HAVE_TDM
  // Stage A tile (16x128 -> LDS stride 132) with the Tensor Data Mover:
  // pad_interval=6 -> every 128 DWORDs, pad_amount=3 -> +4 DWORDs.
  if (w == 0) {
    tdm_load_2d(&x[(size_t)m0 * INsz], (unsigned)(__SIZE_TYPE__)(void*)xT,
                INsz, 16, INsz, /*padInt=*/6, /*padAmt=*/3);
    __builtin_amdgcn_s_wait_tensorcnt(0);
  }
#else
#pragma unroll
  for (int i = 0; i < 2; ++i) {
    int v4 = tid * 2 + i;
    int r = v4 >> 5, k4 = (v4 & 31) * 4;
    const float4 s = *(const float4*)&x[(size_t)(m0 + r) * INsz + k4];
    *(float4*)&xT[r * 132 + k4] = s;
  }
#endif
  __syncthreads();

  v8f acc = {0.f, 0.f, 0.f, 0.f, 0.f, 0.f, 0.f, 0.f};
  const float* Wr = &W[(size_t)(n0 + lm) * INsz];
  for (int kk = 0; kk < 32; ++kk) {
    const int k0 = kk * 4 + cs;
    // A: lane half selects K-pair {k0,k0+1} / {k0+2,k0+3}, row M = lane%16
    v2f a = *(const v2f*)&xT[lm * 132 + k0];
    // B[k][n] = W[n0+n][k] -> contiguous pair along K
    v2f b = *(const v2f*)&Wr[k0];
    acc = wmma_k4(a, b, acc);
  }
#pragma unroll
  for (int r = 0; r < 8; ++r)
    iProj[(size_t)(m0 + r + hl * 8) * 512 + n0 + lm] = acc[r];
}

// ---------------------------------------------------------------------
// K2: elementwise EMA scan for key_state/key/key_trace (no mem recurrence)
// ---------------------------------------------------------------------
__global__ __launch_bounds__(256) void key_scan(const float* __restrict__ iProj,
                                                float* __restrict__ keysOut,
                                                float* __restrict__ ktBuf) {
  const int b = blockIdx.x, h = threadIdx.x;
  float ks = 0.f, kt = 0.f;
  for (int t = 0; t < Tsz; ++t) {
    const size_t row = (size_t)((b << 7) + t);
    float ik = iProj[row * 512 + h];
    ks = ALPHA_C * ks + OMA_C * ik;
    float key = fmaxf(ks, 0.f);
    kt = DT_C * kt + OMD_C * key;
    keysOut[row * Hsz + h] = key;  // also the "keys_seq" output
    ktBuf[row * Hsz + h] = kt;
  }
}

// ---------------------------------------------------------------------
// K3: chunked recurrent scan. 1 block per batch element, mem in LDS.
// LDS layout (floats):
//   memS [256][260]  @ 0       (266,240 B)  row-pad 260 -> conflict-free WMMA A
//   keyS [16][260]   @ 66560   key chunk, [n][j]        (TDM-staged)
//   ktS  [16][260]   @ 70720   key-trace chunk, [s][j]  (TDM-staged)
//   mvtS [256][18]   @ 74880   M result (cols 0..15), later LR*VT for update
//   gS   [16][17]    @ 79488   Gram
//   valS [256]       @ 79760
//   partS[256]       @ 80016
//   viB  [4][64]     @ 80272
// total 80528 floats = 322,112 B  (<= 320 KB per WGP)
// ---------------------------------------------------------------------
#define MEM_ST 260
#define KCH_ST 260
#define MVT_ST 18
#define OFF_KEY 66560
#define OFF_KT  70720
#define OFF_MVT 74880
#define OFF_G   79488
#define OFF_VAL 79760
#define OFF_PART 80016
#define OFF_VIB 80272
#define LDS_FLOATS 80528

__global__ __launch_bounds__(256, 1) void scan_kernel(
    const float* __restrict__ iProj, const float* __restrict__ keysOut,
    const float* __restrict__ ktBuf, const float* __restrict__ W_vei,
    const float* __restrict__ W_ive, float* __restrict__ outMem,
    float* __restrict__ outVals) {
  extern __shared__ float sm[];
  float* memS = sm;
  float* keyS = sm + OFF_KEY;
  float* ktS = sm + OFF_KT;
  float* mvtS = sm + OFF_MVT;
  float* gS = sm + OFF_G;
  float* valS = sm + OFF_VAL;
  float* partS = sm + OFF_PART;
  float* viB = sm + OFF_VIB;

  const int b = blockIdx.x, tid = threadIdx.x, lane = tid & 31, w = tid >> 5;
  const int lm = lane & 15, hl = lane >> 4, cs = hl * 2;
  const int cI = tid & 63, qI = tid >> 6;

  // zero mem + vi buffer
  for (int i = tid; i < 256 * MEM_ST; i += 256) memS[i] = 0.f;
  viB[tid] = 0.f;  // 4*64 = 256 entries

  // register-resident inhibition weights (fully unrolled indexing only)
  float Wive[64];  // W_ive row h = tid  (vio = W_ive[h,:] . vi_read)
  const float* wiveRow = &W_ive[(size_t)tid * 64];
#pragma unroll
  for (int c = 0; c < 64; ++c) Wive[c] = wiveRow[c];
  float Wvei[64];  // thread (cI,qI): W_vei[cI][qI*64 + j]
  const float* wveiRow = &W_vei[(size_t)cI * 256 + qI * 64];
#pragma unroll
  for (int j = 0; j < 64; ++j) Wvei[j] = wveiRow[j];

  float vs = 0.f, vtr = 0.f, vis = 0.f;
  float vtreg[16];
#pragma unroll
  for (int s = 0; s < 16; ++s) vtreg[s] = 0.f;

  __syncthreads();

  for (int ch = 0; ch < 8; ++ch) {
    const int t0 = ch * 16;
    // ---- phase 1: stage key / key-trace chunk into LDS ----
#if HAVE_TDM
    // Two TDM descriptors per chunk: 16x256 f32 tiles, hardware-padded to
    // LDS row stride 260 (pad_interval=7: every 256 DW; pad_amount=3: +4 DW).
    if (w == 0) {
      const size_t gOff = (size_t)((b << 7) + t0) * Hsz;
      tdm_load_2d(&keysOut[gOff], (unsigned)(__SIZE_TYPE__)(void*)keyS,
                  Hsz, 16, Hsz, /*padInt=*/7, /*padAmt=*/3);
      tdm_load_2d(&ktBuf[gOff], (unsigned)(__SIZE_TYPE__)(void*)ktS,
                  Hsz, 16, Hsz, /*padInt=*/7, /*padAmt=*/3);
      __builtin_amdgcn_s_wait_tensorcnt(0);
    }
#else
    for (int idx = tid; idx < 16 * 256; idx += 256) {
      const int n = idx >> 8, j = idx & 255;
      const size_t row = (size_t)((b << 7) + t0 + n);
      keyS[n * KCH_ST + j] = keysOut[row * Hsz + j];
      ktS[n * KCH_ST + j] = ktBuf[row * Hsz + j];
    }
#endif
    __syncthreads();

    // ---- phase 2: WMMA  M = mem @ Keys  (16 M-tiles over 8 waves) ----
    for (int mt = w; mt < 16; mt += 8) {
      const int m0 = mt * 16;
      v8f acc = {0.f, 0.f, 0.f, 0.f, 0.f, 0.f, 0.f, 0.f};
      for (int kk = 0; kk < 64; ++kk) {
        const int k0 = kk * 4 + cs;
        v2f a = *(const v2f*)&memS[(m0 + lm) * MEM_ST + k0];   // A[m][k]
        v2f bb = *(const v2f*)&keyS[lm * KCH_ST + k0];         // B[k][n]=key_n[k]
        acc = wmma_k4(a, bb, acc);
      }
#pragma unroll
      for (int r = 0; r < 8; ++r)
        mvtS[(m0 + r + hl * 8) * MVT_ST + lm] = acc[r];
    }
    // Gram G[s][n] = kt_s . key_n  (wave 0)
    if (w == 0) {
      v8f g = {0.f, 0.f, 0.f, 0.f, 0.f, 0.f, 0.f, 0.f};
      for (int kk = 0; kk < 64; ++kk) {
        const int k0 = kk * 4 + cs;
        v2f a = *(const v2f*)&ktS[lm * KCH_ST + k0];
        v2f bb = *(const v2f*)&keyS[lm * KCH_ST + k0];
        g = wmma_k4(a, bb, g);
      }
#pragma unroll
      for (int r = 0; r < 8; ++r) gS[(r + hl * 8) * 17 + lm] = g[r];
    }
    __syncthreads();

    // ---- phase 3: 16 sequential steps (VALU) ----
    for (int n = 0; n < 16; ++n) {
      const int t = t0 + n, slot = t & 3;
      // vio[h] = W_ive[h,:] . vi_buffer[slot]   (broadcast LDS reads)
      float vio = 0.f;
#pragma unroll
      for (int c = 0; c < 64; ++c) vio = fmaf(Wive[c], viB[slot * 64 + c], vio);
      // lazy rank-1 correction: sum_{s<n} vt_s[h] * G[s][n]
      float corr = 0.f;
#pragma unroll
      for (int s = 0; s < 16; ++s) {
        float gv = gS[s * 17 + n];
        corr += (s < n) ? vtreg[s] * gv : 0.f;
      }
      const float ikv = RECALL_C * fmaf(LR_C, corr, mvtS[tid * MVT_ST + n]);
      const size_t row = (size_t)((b << 7) + t);
      __builtin_prefetch(&iProj[(row + 1) * 512 + 256 + tid], 0, 0);
      const float ivv = iProj[row * 512 + 256 + tid];
      vs = ALPHA_C * vs + OMA_C * (ivv + ikv + vio);
      const float val = fmaxf(vs, 0.f);
      valS[tid] = val;
      vtr = DT_C * vtr + OMD_C * val;
#pragma unroll
      for (int s = 0; s < 16; ++s)
        if (s == n) vtreg[s] = vtr;  // const-indexed -> stays in VGPRs
      outVals[row * Hsz + tid] = val;
      __syncthreads();
      // vii[c] = sum_h val[h] * W_vei[c][h]  (4-way split reduction)
      float p = 0.f;
      const int hb = qI * 64;
#pragma unroll
      for (int j = 0; j < 64; ++j) p = fmaf(valS[hb + j], Wvei[j], p);
      partS[qI * 64 + cI] = p;
      __syncthreads();
      if (tid < 64) {
        const float vii =
            partS[tid] + partS[64 + tid] + partS[128 + tid] + partS[192 + tid];
        vis = ALPHA_C * vis + OMA_C * vii;
        viB[slot * 64 + tid] = fmaxf(vis, 0.f);
      }
      __syncthreads();
    }

    // ---- phase 4: mem += LR * VT @ KT^T   (WMMA rank-16, C-accumulate) ----
#pragma unroll
    for (int s = 0; s < 16; ++s) mvtS[tid * MVT_ST + s] = LR_C * vtreg[s];
    __syncthreads();
    for (int tile = w; tile < 256; tile += 8) {
      const int m0 = (tile >> 4) * 16, n0 = (tile & 15) * 16;
      v8f acc;
#pragma unroll
      for (int r = 0; r < 8; ++r)
        acc[r] = memS[(m0 + r + hl * 8) * MEM_ST + n0 + lm];
#pragma unroll
      for (int kk = 0; kk < 4; ++kk) {
        const int k0 = kk * 4 + cs;
        v2f a = *(const v2f*)&mvtS[(m0 + lm) * MVT_ST + k0];  // A[m][s]=LR*vt_s[m]
        v2f bb;                                                // B[s][n]=kt_s[n0+n]
        bb.x = ktS[k0 * KCH_ST + n0 + lm];
        bb.y = ktS[(k0 + 1) * KCH_ST + n0 + lm];
        acc = wmma_k4(a, bb, acc);
      }
#pragma unroll
      for (int r = 0; r < 8; ++r)
        memS[(m0 + r + hl * 8) * MEM_ST + n0 + lm] = acc[r];
    }
    __syncthreads();
  }

  // ---- final writeback of mem ----
  for (int r = 0; r < 256; ++r)
    outMem[(size_t)b * 65536 + r * 256 + tid] = memS[r * MEM_ST + tid];
}

// ---------------------------------------------------------------------
extern "C" void kernel_launch(void* const* d_in, const int* in_sizes, int n_in,
                              void* d_out, int out_size, void* d_ws,
                              size_t ws_size, hipStream_t stream) {
  const float* x = (const float*)d_in[0];      // (64,128,128)
  const float* W = (const float*)d_in[1];      // (512,128)
  const float* W_vei = (const float*)d_in[2];  // (64,256)
  const float* W_ive = (const float*)d_in[3];  // (256,64)
  float* out = (float*)d_out;
  float* ws = (float*)d_ws;

  float* iProj = ws;                 // 8192*512 floats (16 MB)
  float* ktBuf = ws + 4194304;       // 64*128*256 floats (8 MB)
  float* outMem = out;               // B*H*H
  float* keysOut = out + 4194304;    // B*T*H
  float* outVals = out + 6291456;    // B*T*H

  proj_gemm<<<dim3(512, 4), 256, 0, stream>>>(x, W, iProj);
  key_scan<<<Bsz, 256, 0, stream>>>(iProj, keysOut, ktBuf);

  const size_t shmem = (size_t)LDS_FLOATS * sizeof(float);  // 322,112 B
  (void)hipFuncSetAttribute((const void*)scan_kernel,
                            hipFuncAttributeMaxDynamicSharedMemorySize,
                            (int)shmem);
  scan_kernel<<<Bsz, 256, shmem, stream>>>(iProj, keysOut, ktBuf, W_vei, W_ive,
                                           outMem, outVals);
}